// TrajPredictor_19980187861264
// MI455X (gfx1250) — compile-verified
//
#include <hip/hip_runtime.h>
#include <cmath>

typedef float v2f __attribute__((ext_vector_type(2)));
typedef float v8f __attribute__((ext_vector_type(8)));

constexpr int Hdim  = 128;   // hidden size
constexpr int TT    = 64;    // encoder steps
constexpr int PLEN  = 32;    // decoder steps
constexpr int WAVES = 4;     // waves per workgroup (each wave = one 16-row batch tile)
constexpr int NJ    = 8;     // 128/16 n-tiles
constexpr int NK    = 32;    // 128/4  k-chunks
// packed B operand: [NK][NJ][32 lanes][2] = 16384 floats (= H*H)
constexpr int BPACK = NK * NJ * 32 * 2;
// packed A operand (h state) per wave: [NK][32 lanes][2] = 2048 floats (= 16*H)
constexpr int APACK = NK * 32 * 2;

extern __shared__ float smem[];

// flat index of h[m][k] (m=row 0..15, k=0..127) in the A-fragment-packed layout
__device__ __forceinline__ int a_idx(int m, int k) {
    // consumer lane = (((k>>1)&1)<<4) | m ; vgpr p = k&1 ; chunk kk = k>>2
    return (k >> 2) * 64 + (((((k >> 1) & 1) << 4) + m) << 1) + (k & 1);
}

// branch-free tanh: 1 - 2/(exp2(2*log2(e)*x) + 1)
// single v_exp_f32 + v_rcp_f32, no EXEC-mask churn, saturates correctly at +/-1
__device__ __forceinline__ float fast_tanh(float x) {
    float e = __builtin_amdgcn_exp2f(x * 2.88539008177792681472f); // 2*log2(e)
    return 1.0f - 2.0f * __builtin_amdgcn_rcpf(e + 1.0f);
}

__global__ __launch_bounds__(WAVES * 32)
void traj_rnn_kernel(const float* __restrict__ x,
                     const float* __restrict__ eWih, const float* __restrict__ eWhh,
                     const float* __restrict__ ebih, const float* __restrict__ ebhh,
                     const float* __restrict__ dWih, const float* __restrict__ dWhh,
                     const float* __restrict__ dbih, const float* __restrict__ dbhh,
                     const float* __restrict__ fcW,  const float* __restrict__ fcb,
                     float* __restrict__ out, int B)
{
    // LDS layout
    float* packBe = smem;                    // BPACK  (enc Whh^T, B-fragment order)
    float* packBd = packBe + BPACK;          // BPACK  (dec Whh^T, B-fragment order)
    float* fcWl   = packBd + BPACK;          // 3*Hdim
    float* fcbl   = fcWl + 3 * Hdim;         // 4
    float* hall   = fcbl + 4;                // WAVES * APACK (per-wave h, A-frag order)
    float* iall   = hall + WAVES * APACK;    // WAVES * 16 * 4 (per-wave decoder input)

    const int tid  = threadIdx.x;
    const int wave = tid >> 5;
    const int lane = tid & 31;
    const int half = lane >> 4;   // 0: lanes 0-15, 1: lanes 16-31
    const int ln   = lane & 15;

    // Stage Whh matrices directly in B-fragment-packed order:
    // idx = ((kk*NJ + j)*32 + l)*2 + p  maps to  Whh[n][k], n=j*16+(l&15), k=kk*4+(l>>4)*2+p
    for (int idx = tid; idx < BPACK; idx += blockDim.x) {
        int p  = idx & 1;
        int l  = (idx >> 1) & 31;
        int j  = (idx >> 6) & 7;
        int kk = idx >> 9;
        int n  = j * 16 + (l & 15);
        int k  = kk * 4 + ((l >> 4) << 1) + p;
        packBe[idx] = eWhh[n * Hdim + k];
        packBd[idx] = dWhh[n * Hdim + k];
    }
    for (int idx = tid; idx < 3 * Hdim; idx += blockDim.x) fcWl[idx] = fcW[idx];
    if (tid < 3) fcbl[tid] = fcb[tid];
    __syncthreads();

    const int tile = blockIdx.x * WAVES + wave;
    const int b0   = tile * 16;
    if (b0 >= B) return;   // uniform per wave; no barriers after this point

    float* hA = hall + wave * APACK;
    float* il = iall + wave * (16 * 4);

    // Per-lane constants: for C/D frag j this lane's column is j*16 + ln
    float bse[NJ], bsd[NJ], wie[NJ][3], wid[NJ][3];
#pragma unroll
    for (int j = 0; j < NJ; ++j) {
        int col = j * 16 + ln;
        bse[j] = ebih[col] + ebhh[col];
        bsd[j] = dbih[col] + dbhh[col];
#pragma unroll
        for (int i = 0; i < 3; ++i) {
            wie[j][i] = eWih[col * 3 + i];
            wid[j][i] = dWih[col * 3 + i];
        }
    }

    // h0 = 0
    for (int idx = lane; idx < APACK; idx += 32) hA[idx] = 0.0f;

    v8f c[NJ];

    // ---------------- encoder scan ----------------
    for (int t = 0; t < TT; ++t) {
        float xv[8][3];
#pragma unroll
        for (int r = 0; r < 8; ++r) {
            const float* xp = x + ((size_t)(b0 + r + half * 8) * TT + t) * 3;
#pragma unroll
            for (int i = 0; i < 3; ++i) xv[r][i] = xp[i];
        }
        // C init = bih + bhh + x_t @ Wih^T   (C/D layout: reg r -> row r+half*8, col j*16+ln)
#pragma unroll
        for (int j = 0; j < NJ; ++j)
#pragma unroll
            for (int r = 0; r < 8; ++r)
                c[j][r] = bse[j] + xv[r][0] * wie[j][0] + xv[r][1] * wie[j][1]
                                 + xv[r][2] * wie[j][2];

        // accumulate h @ Whh^T with fp32 WMMA 16x16x4 (all operands: one ds_load_b64 each)
#pragma unroll 4
        for (int kk = 0; kk < NK; ++kk) {
            v2f a = *(const v2f*)&hA[(kk * 32 + lane) * 2];
#pragma unroll
            for (int j = 0; j < NJ; ++j) {
                v2f bf = *(const v2f*)&packBe[((kk * NJ + j) * 32 + lane) * 2];
                c[j] = __builtin_amdgcn_wmma_f32_16x16x4_f32(
                        false, a, false, bf, (short)0, c[j], false, false);
            }
        }
        // h = tanh(pre); store straight into A-fragment-packed layout
#pragma unroll
        for (int j = 0; j < NJ; ++j)
#pragma unroll
            for (int r = 0; r < 8; ++r)
                hA[a_idx(r + half * 8, j * 16 + ln)] = fast_tanh(c[j][r]);
    }

    // ---------------- decoder ----------------
    // dec_in0 = x[:, T-1, :]
    if (lane < 16) {
#pragma unroll
        for (int i = 0; i < 3; ++i)
            il[lane * 4 + i] = x[((size_t)(b0 + lane) * TT + (TT - 1)) * 3 + i];
    }

    for (int p = 0; p < PLEN; ++p) {
        float xv[8][3];
#pragma unroll
        for (int r = 0; r < 8; ++r)
#pragma unroll
            for (int i = 0; i < 3; ++i) xv[r][i] = il[(r + half * 8) * 4 + i];
#pragma unroll
        for (int j = 0; j < NJ; ++j)
#pragma unroll
            for (int r = 0; r < 8; ++r)
                c[j][r] = bsd[j] + xv[r][0] * wid[j][0] + xv[r][1] * wid[j][1]
                                 + xv[r][2] * wid[j][2];

#pragma unroll 4
        for (int kk = 0; kk < NK; ++kk) {
            v2f a = *(const v2f*)&hA[(kk * 32 + lane) * 2];
#pragma unroll
            for (int j = 0; j < NJ; ++j) {
                v2f bf = *(const v2f*)&packBd[((kk * NJ + j) * 32 + lane) * 2];
                c[j] = __builtin_amdgcn_wmma_f32_16x16x4_f32(
                        false, a, false, bf, (short)0, c[j], false, false);
            }
        }
#pragma unroll
        for (int j = 0; j < NJ; ++j)
#pragma unroll
            for (int r = 0; r < 8; ++r)
                hA[a_idx(r + half * 8, j * 16 + ln)] = fast_tanh(c[j][r]);

        // pred = h_new @ fc_W^T + fc_b ; feed back as next input and store output
        if (lane < 16) {
            float a0 = fcbl[0], a1 = fcbl[1], a2 = fcbl[2];
#pragma unroll 8
            for (int k = 0; k < Hdim; ++k) {
                float hv = hA[a_idx(lane, k)];
                a0 += hv * fcWl[0 * Hdim + k];
                a1 += hv * fcWl[1 * Hdim + k];
                a2 += hv * fcWl[2 * Hdim + k];
            }
            size_t ob = ((size_t)(b0 + lane) * PLEN + p) * 3;
            out[ob + 0] = a0; out[ob + 1] = a1; out[ob + 2] = a2;
            il[lane * 4 + 0] = a0; il[lane * 4 + 1] = a1; il[lane * 4 + 2] = a2;
        }
    }
}

extern "C" void kernel_launch(void* const* d_in, const int* in_sizes, int n_in,
                              void* d_out, int out_size, void* d_ws, size_t ws_size,
                              hipStream_t stream) {
    (void)n_in; (void)out_size; (void)d_ws; (void)ws_size;
    const float* x    = (const float*)d_in[0];
    const float* eWih = (const float*)d_in[1];
    const float* eWhh = (const float*)d_in[2];
    const float* ebih = (const float*)d_in[3];
    const float* ebhh = (const float*)d_in[4];
    const float* dWih = (const float*)d_in[5];
    const float* dWhh = (const float*)d_in[6];
    const float* dbih = (const float*)d_in[7];
    const float* dbhh = (const float*)d_in[8];
    const float* fcW  = (const float*)d_in[9];
    const float* fcb  = (const float*)d_in[10];
    float* out = (float*)d_out;

    int B = in_sizes[0] / (TT * 3);           // 4096
    int tiles  = (B + 15) / 16;               // 256
    int blocks = (tiles + WAVES - 1) / WAVES; // 64

    size_t shmem = (size_t)(2 * BPACK + 3 * Hdim + 4
                            + WAVES * APACK + WAVES * 16 * 4) * sizeof(float);

    traj_rnn_kernel<<<blocks, WAVES * 32, shmem, stream>>>(
        x, eWih, eWhh, ebih, ebhh, dWih, dWhh, dbih, dbhh, fcW, fcb, out, B);
}